// Quantization_Layer_59176059404482
// MI455X (gfx1250) — compile-verified
//
#include <hip/hip_runtime.h>
#include <math.h>

// Quantization layer forward: y = c[argmin_k (x - c[k])^2], idx = argmin.
// c[k] = 6*sigmoid(centroids[k]), 8 centroids.
// Memory-bound streaming kernel: ~308 MB traffic -> ~13us at 23.3 TB/s.
// Outputs: d_out[0..n)   = y   (float)
//          d_out[n..2n)  = idx (written as float values 0..7)

typedef float v4f __attribute__((ext_vector_type(4)));
typedef int   v4i __attribute__((ext_vector_type(4)));

// Builtin parameter types (revealed by round-1 diagnostics):
// (int4 addrspace(1)*, int4 addrspace(3)*, imm offset, imm cpol)
typedef __attribute__((address_space(1))) v4i* gas_v4i_ptr;
typedef __attribute__((address_space(3))) v4i* las_v4i_ptr;

#define NUM_C 8
#define TPB   256
#define VPT   8        // float4 chunks per thread -> 32KB LDS tile per block

#if defined(__AMDGCN__) && defined(__gfx1250__) && \
    __has_builtin(__builtin_amdgcn_global_load_async_to_lds_b128) && \
    __has_builtin(__builtin_amdgcn_s_wait_asynccnt)
#define USE_ASYNC_LDS 1
#else
#define USE_ASYNC_LDS 0
#endif

// Activated centroids, computed once per launch by the setup kernel.
// 16B-aligned so the hot kernel can fetch it as two b128 loads.
__device__ __align__(16) float g_cvals[NUM_C];

__device__ __forceinline__ void classify(float xs, const float* c,
                                         float& y, float& idx) {
    float d0 = xs - c[0];
    float bd = d0 * d0;   // best squared distance
    float bc = c[0];      // best centroid value
    float bi = 0.0f;      // best index (as float)
#pragma unroll
    for (int k = 1; k < NUM_C; ++k) {
        float d  = xs - c[k];
        float d2 = d * d;
        bool lt = d2 < bd;   // strict <: first-min tie-break like jnp.argmin
        bd = lt ? d2 : bd;
        bc = lt ? c[k] : bc;
        bi = lt ? (float)k : bi;
    }
    y = bc;
    idx = bi;
}

__device__ __forceinline__ void process_store(v4f xsv, const float* c,
                                              float* __restrict__ y_out,
                                              float* __restrict__ idx_out,
                                              long long i) {
    v4f yv, iv;
#pragma unroll
    for (int j = 0; j < 4; ++j) {
        float yy, ii;
        classify(xsv[j], c, yy, ii);
        yv[j] = yy;
        iv[j] = ii;
    }
    // Streaming stores: outputs are write-once; bypass caches.
    __builtin_nontemporal_store(yv, (v4f*)y_out + i);
    __builtin_nontemporal_store(iv, (v4f*)idx_out + i);
}

// Hot kernel (defined first so it leads the device assembly): pure vector
// stream over nv4 float4 chunks (requires n%4==0).
__global__ __launch_bounds__(TPB) void quant_vec_kernel(
    const float* __restrict__ x,
    float* __restrict__ y_out,
    float* __restrict__ idx_out,
    long long nv4)
{
#if USE_ASYNC_LDS
    __shared__ v4f tile[VPT * TPB];
#endif
    // Centroid table: two 128-bit cached loads.
    v4f clo = *(const v4f*)&g_cvals[0];
    v4f chi = *(const v4f*)&g_cvals[4];
    float c[NUM_C];
#pragma unroll
    for (int k = 0; k < 4; ++k) { c[k] = clo[k]; c[k + 4] = chi[k]; }

    const int t = threadIdx.x;
    long long blockBase = (long long)blockIdx.x * (VPT * TPB);
    const v4f* xv4 = (const v4f*)x;

    if (blockBase + (long long)(VPT * TPB) <= nv4) {
        // ---- Fast path: full block, branch-free straight-line stream. ----
        const v4f* bp = xv4 + blockBase;   // uniform base -> SGPR + voffset
        v4f xs[VPT];
#if USE_ASYNC_LDS
        // CDNA5 async global->LDS path: VPT back-to-back 128-bit async
        // loads, then a single ASYNCcnt wait (VPT-deep MLP). Each lane
        // fills its own LDS slot, so this wave's counter reaching 0 is
        // sufficient -- no barrier needed.
#pragma unroll
        for (int k = 0; k < VPT; ++k)
            __builtin_amdgcn_global_load_async_to_lds_b128(
                (gas_v4i_ptr)(v4i*)(bp + (k * TPB + t)),
                (las_v4i_ptr)(v4i*)&tile[k * TPB + t], 0, 0);
        __builtin_amdgcn_s_wait_asynccnt(0);
#pragma unroll
        for (int k = 0; k < VPT; ++k) xs[k] = tile[k * TPB + t];
#else
#pragma unroll
        for (int k = 0; k < VPT; ++k)
            xs[k] = __builtin_nontemporal_load(bp + (k * TPB + t));
#endif
#pragma unroll
        for (int k = 0; k < VPT; ++k)
            process_store(xs[k], c, y_out, idx_out, blockBase + k * TPB + t);
    } else {
        // ---- Cold path: partial last block, per-chunk guards. ----
#pragma unroll
        for (int k = 0; k < VPT; ++k) {
            long long i = blockBase + k * TPB + t;
            if (i < nv4) {
                v4f xv = __builtin_nontemporal_load(xv4 + i);
                process_store(xv, c, y_out, idx_out, i);
            }
        }
    }
}

__global__ void centroid_setup_kernel(const float* __restrict__ centroids) {
    int k = threadIdx.x;
    if (k < NUM_C) {
        // fp64 so decision boundaries (midpoints) match the reference <1 ulp.
        double v = (double)centroids[k];
        g_cvals[k] = (float)(6.0 / (1.0 + exp(-v)));
    }
}

// Scalar tail for n % 4 != 0 (not hit for this problem's shapes).
__global__ void quant_tail_kernel(const float* __restrict__ x,
                                  float* __restrict__ y_out,
                                  float* __restrict__ idx_out,
                                  long long start, long long n)
{
    long long e = start + (long long)blockIdx.x * blockDim.x + threadIdx.x;
    if (e < n) {
        float c[NUM_C];
#pragma unroll
        for (int k = 0; k < NUM_C; ++k) c[k] = g_cvals[k];
        float yv, iv;
        classify(x[e], c, yv, iv);
        y_out[e]   = yv;
        idx_out[e] = iv;
    }
}

extern "C" void kernel_launch(void* const* d_in, const int* in_sizes, int n_in,
                              void* d_out, int out_size, void* d_ws, size_t ws_size,
                              hipStream_t stream) {
    (void)n_in; (void)out_size; (void)d_ws; (void)ws_size;

    const float* x         = (const float*)d_in[0];   // [64,56,56,128] f32
    const float* centroids = (const float*)d_in[1];   // [1,8] f32

    long long n = (long long)in_sizes[0];
    float* y   = (float*)d_out;        // first n elements: y
    float* idx = (float*)d_out + n;    // next n elements: idx (as float)

    centroid_setup_kernel<<<1, 32, 0, stream>>>(centroids);

    long long nv4 = n >> 2;            // full float4 chunks
    if (nv4 > 0) {
        long long perBlock = (long long)VPT * TPB;
        int blocks = (int)((nv4 + perBlock - 1) / perBlock);
        quant_vec_kernel<<<blocks, TPB, 0, stream>>>(x, y, idx, nv4);
    }
    long long tail = n - (nv4 << 2);
    if (tail > 0) {
        quant_tail_kernel<<<1, 64, 0, stream>>>(x, y, idx, nv4 << 2, n);
    }
}